// TransformerDecoder_81097572483792
// MI455X (gfx1250) — compile-verified
//
#include <hip/hip_runtime.h>
#include <hip/hip_bf16.h>

// =====================================================================
// MI455X (gfx1250, wave32) implementation of the 3D decoder + Mamba net.
//
//  * All channel contractions (3D convs incl. transposed convs, Mamba
//    projections) use V_WMMA_F32_16X16X4_F32 (fp32 = reference fidelity;
//    matrix work is L2-resident at 192MB; net is latency-bound by the
//    sequential Mamba scans).
//  * Each wave owns a 32x32 output tile = 2x2 WMMA sub-tiles: per K-step
//    2 A-pairs + 2 B-pairs feed 4 independent WMMAs (halves operand
//    traffic per WMMA and hides the XDL RAW latency).
//  * Inner loops are BRANCH-FREE: M/N edge tiles handled by pointer
//    clamping (garbage only pollutes never-stored rows/cols of D);
//    conv padding taps use a stride-0 pointer into a zeroed scratch
//    line; only the K%4 tail WMMA uses zero-fill masking.
//  * Hot-loop pointers are individually named scalars (NOT arrays) so
//    clang's addrspace inference emits global_load_b64 (LOADcnt only),
//    not flat_load (LOADcnt+DScnt+aperture checks).
//  * Bias + BN(eval) + ReLU + residual-add fused into conv epilogues.
//  * Mamba selective scan: one thread per (batch, channel), d_state in
//    registers (templated).
//
// Input ordering assumption: d_in[0] = x, then params flattened with
// JAX pytree semantics (dict keys sorted, lists/tuples in order).
// =====================================================================

typedef float v2f __attribute__((ext_vector_type(2)));
typedef float v8f __attribute__((ext_vector_type(8)));

static __device__ __forceinline__ v8f wmma4(v2f a, v2f b, v8f c) {
  // D = A(16x4) * B(4x16) + C(16x16), fp32
  return __builtin_amdgcn_wmma_f32_16x16x4_f32(false, a, false, b,
                                               (short)0, c, false, false);
}

// ---------------------------------------------------------------------
// Generic GEMM: out[M,N] = A[M,K] * B[N,K]^T (+bias[n]) (+activation)
// act: 0 = none, 1 = relu, 2 = softplus.  32x32 tile per wave.
// ---------------------------------------------------------------------
__global__ __launch_bounds__(128) void gemm_wmma_kernel(
    const float* __restrict__ A, int lda,
    const float* __restrict__ Bw, int ldb,
    const float* __restrict__ bias,
    float* __restrict__ out, int ldo,
    int M, int N, int K, int act) {
  const int wave = threadIdx.x >> 5;
  const int lane = threadIdx.x & 31;
  const int l16 = lane & 15;
  const int hi = lane >> 4;  // which K-pair (loads) / which M-half (store)
  const long mBase = ((long)blockIdx.x * 4 + wave) * 32;
  const int nBase = blockIdx.y * 32;

  // Clamped row/col pointers (edge garbage lands in never-stored D cells).
  const long m0 = mBase + l16;
  const long m1 = mBase + 16 + l16;
  const long mCl0 = (m0 < M) ? m0 : (long)(M - 1);
  const long mCl1 = (m1 < M) ? m1 : (long)(M - 1);
  const int n0 = nBase + l16;
  const int n1 = nBase + 16 + l16;
  const int nCl0 = (n0 < N) ? n0 : (N - 1);
  const int nCl1 = (n1 < N) ? n1 : (N - 1);

  const float* aP0 = A + mCl0 * (long)lda + hi * 2;
  const float* aP1 = A + mCl1 * (long)lda + hi * 2;
  const float* bP0 = Bw + (long)nCl0 * ldb + hi * 2;
  const float* bP1 = Bw + (long)nCl1 * ldb + hi * 2;

  v8f acc[2][2];
#pragma unroll
  for (int mi = 0; mi < 2; ++mi)
#pragma unroll
    for (int nj = 0; nj < 2; ++nj)
      acc[mi][nj] = {0.f, 0.f, 0.f, 0.f, 0.f, 0.f, 0.f, 0.f};

  const int Kmain = K & ~3;
  for (int k0 = 0; k0 < Kmain; k0 += 4) {
    v2f a0, a1, b0, b1;
    a0.x = aP0[0]; a0.y = aP0[1];
    a1.x = aP1[0]; a1.y = aP1[1];
    b0.x = bP0[0]; b0.y = bP0[1];
    b1.x = bP1[0]; b1.y = bP1[1];
    aP0 += 4; aP1 += 4; bP0 += 4; bP1 += 4;
    acc[0][0] = wmma4(a0, b0, acc[0][0]);
    acc[0][1] = wmma4(a0, b1, acc[0][1]);
    acc[1][0] = wmma4(a1, b0, acc[1][0]);
    acc[1][1] = wmma4(a1, b1, acc[1][1]);
  }
  if (K & 3) {  // K-tail: only K overflow pollutes valid outputs -> mask
    const int k = Kmain + hi * 2;
    const bool k0ok = (k < K), k1ok = ((k + 1) < K);
    const float* aR0 = A + mCl0 * (long)lda;
    const float* aR1 = A + mCl1 * (long)lda;
    const float* bR0 = Bw + (long)nCl0 * ldb;
    const float* bR1 = Bw + (long)nCl1 * ldb;
    v2f a0, a1, b0, b1;
    a0.x = k0ok ? aR0[k] : 0.f; a0.y = k1ok ? aR0[k + 1] : 0.f;
    a1.x = k0ok ? aR1[k] : 0.f; a1.y = k1ok ? aR1[k + 1] : 0.f;
    b0.x = k0ok ? bR0[k] : 0.f; b0.y = k1ok ? bR0[k + 1] : 0.f;
    b1.x = k0ok ? bR1[k] : 0.f; b1.y = k1ok ? bR1[k + 1] : 0.f;
    acc[0][0] = wmma4(a0, b0, acc[0][0]);
    acc[0][1] = wmma4(a0, b1, acc[0][1]);
    acc[1][0] = wmma4(a1, b0, acc[1][0]);
    acc[1][1] = wmma4(a1, b1, acc[1][1]);
  }

#pragma unroll
  for (int nj = 0; nj < 2; ++nj) {
    const int n = nBase + nj * 16 + l16;
    if (n < N) {
      const float bb = bias ? bias[n] : 0.f;
#pragma unroll
      for (int mi = 0; mi < 2; ++mi) {
#pragma unroll
        for (int i = 0; i < 8; ++i) {
          const long m = mBase + mi * 16 + hi * 8 + i;
          if (m < M) {
            float v = acc[mi][nj][i] + bb;
            if (act == 1) v = fmaxf(v, 0.f);
            else if (act == 2) v = (v > 20.f) ? v : log1pf(__expf(v));
            out[m * (long)ldo + n] = v;
          }
        }
      }
    }
  }
}

// ---------------------------------------------------------------------
// Generic conv3d (NCDHW) as implicit GEMM over Cin, with lhs-dilation
// (transposed conv) and fused bias + BN(eval) + ReLU + residual.
// wt_flip=1: weight array is (Cin, Cout, KD,KH,KW), taps flipped.
// 32 output positions x 32 output channels per wave (2x2 WMMA tiles).
// ---------------------------------------------------------------------
struct Conv3dArgs {
  const float* in;    // N,Cin,ID,IH,IW
  const float* w;
  const float* bias;  // Cout or null
  const float* bn_g;  // BN gamma or null
  const float* bn_b;  // BN beta  or null
  const float* res;   // residual (same shape as out) or null
  const float* zbuf;  // zeroed scratch line (stride-0 dummy source)
  float* out;         // N,Cout,OD,OH,OW
  int N, Cin, ID, IH, IW;
  int Cout, OD, OH, OW;
  int KD, KH, KW;
  int sd, sh, sw;
  int pd, ph, pw;
  int dd, dh, dw;  // lhs dilation
  int wt_flip;
  int act;  // 0 none, 1 relu
};

__global__ __launch_bounds__(128) void conv3d_wmma_kernel(Conv3dArgs p) {
  const int wave = threadIdx.x >> 5;
  const int lane = threadIdx.x & 31;
  const int l16 = lane & 15;
  const int hi = lane >> 4;
  const int spOut = p.OD * p.OH * p.OW;
  const long Mtot = (long)p.N * spOut;
  const long mBase = ((long)blockIdx.x * 4 + wave) * 32;
  const int nBase = blockIdx.y * 32;

  // Two A-row position decompositions (clamped), named scalars.
  const long mm0 = mBase + l16;
  const long mm1 = mBase + 16 + l16;
  const bool av0 = (mm0 < Mtot);
  const bool av1 = (mm1 < Mtot);
  int ow0, oh0, od0, ni0, ow1, oh1, od1, ni1;
  {
    long r = av0 ? mm0 : (Mtot - 1);
    ow0 = (int)(r % p.OW); r /= p.OW;
    oh0 = (int)(r % p.OH); r /= p.OH;
    od0 = (int)(r % p.OD); r /= p.OD;
    ni0 = (int)r;
  }
  {
    long r = av1 ? mm1 : (Mtot - 1);
    ow1 = (int)(r % p.OW); r /= p.OW;
    oh1 = (int)(r % p.OH); r /= p.OH;
    od1 = (int)(r % p.OD); r /= p.OD;
    ni1 = (int)r;
  }
  const int c0i = nBase + l16;
  const int c1i = nBase + 16 + l16;
  const int coCl0 = (c0i < p.Cout) ? c0i : (p.Cout - 1);
  const int coCl1 = (c1i < p.Cout) ? c1i : (p.Cout - 1);

  const long inCst = (long)p.ID * p.IH * p.IW;  // input channel stride
  const long wCst = p.wt_flip ? (long)p.Cout * p.KD * p.KH * p.KW
                              : (long)p.KD * p.KH * p.KW;

  v8f acc[2][2];
#pragma unroll
  for (int mi = 0; mi < 2; ++mi)
#pragma unroll
    for (int nj = 0; nj < 2; ++nj)
      acc[mi][nj] = {0.f, 0.f, 0.f, 0.f, 0.f, 0.f, 0.f, 0.f};

  const int Kmain = p.Cin & ~3;
  const int rem = p.Cin & 3;

  for (int kd = 0; kd < p.KD; ++kd)
    for (int kh = 0; kh < p.KH; ++kh)
      for (int kw = 0; kw < p.KW; ++kw) {
        // Per-row tap validity; stride-0 zero-line trick keeps the inner
        // loop branch-free.  All scalars by name (no pointer arrays).
        bool vt0, vt1;
        long aIdx0, aIdx1;
        {
          const int td = od0 * p.sd - p.pd + kd;
          const int th = oh0 * p.sh - p.ph + kh;
          const int tw = ow0 * p.sw - p.pw + kw;
          bool v = av0 && (td >= 0) && (th >= 0) && (tw >= 0) &&
                   (td % p.dd == 0) && (th % p.dh == 0) && (tw % p.dw == 0);
          const int id = td / p.dd, ih = th / p.dh, iw = tw / p.dw;
          v = v && (id < p.ID) && (ih < p.IH) && (iw < p.IW);
          vt0 = v;
          aIdx0 = v ? ((((long)ni0 * p.Cin * p.ID + id) * p.IH + ih) * p.IW +
                       iw)
                    : 0;
        }
        {
          const int td = od1 * p.sd - p.pd + kd;
          const int th = oh1 * p.sh - p.ph + kh;
          const int tw = ow1 * p.sw - p.pw + kw;
          bool v = av1 && (td >= 0) && (th >= 0) && (tw >= 0) &&
                   (td % p.dd == 0) && (th % p.dh == 0) && (tw % p.dw == 0);
          const int id = td / p.dd, ih = th / p.dh, iw = tw / p.dw;
          v = v && (id < p.ID) && (ih < p.IH) && (iw < p.IW);
          vt1 = v;
          aIdx1 = v ? ((((long)ni1 * p.Cin * p.ID + id) * p.IH + ih) * p.IW +
                       iw)
                    : 0;
        }
        const long astep0 = vt0 ? inCst : 0;
        const long astep1 = vt1 ? inCst : 0;
        const float* aP0 = (vt0 ? (p.in + aIdx0) : p.zbuf) + hi * 2 * astep0;
        const float* aP1 = (vt1 ? (p.in + aIdx1) : p.zbuf) + hi * 2 * astep1;

        long wIdx0, wIdx1;
        if (!p.wt_flip) {
          wIdx0 = (((long)coCl0 * p.Cin * p.KD + kd) * p.KH + kh) * p.KW + kw;
          wIdx1 = (((long)coCl1 * p.Cin * p.KD + kd) * p.KH + kh) * p.KW + kw;
        } else {
          const int fd = p.KD - 1 - kd, fh = p.KH - 1 - kh, fw = p.KW - 1 - kw;
          wIdx0 = (((long)coCl0 * p.KD + fd) * p.KH + fh) * p.KW + fw;
          wIdx1 = (((long)coCl1 * p.KD + fd) * p.KH + fh) * p.KW + fw;
        }
        const float* bP0 = p.w + wIdx0 + hi * 2 * wCst;
        const float* bP1 = p.w + wIdx1 + hi * 2 * wCst;

        const long aAdv0 = 4 * astep0, aAdv1 = 4 * astep1;
        const long bAdv = 4 * wCst;
        for (int c0 = 0; c0 < Kmain; c0 += 4) {
          v2f a0, a1, b0, b1;
          a0.x = aP0[0]; a0.y = aP0[astep0];
          a1.x = aP1[0]; a1.y = aP1[astep1];
          b0.x = bP0[0]; b0.y = bP0[wCst];
          b1.x = bP1[0]; b1.y = bP1[wCst];
          aP0 += aAdv0; aP1 += aAdv1; bP0 += bAdv; bP1 += bAdv;
          acc[0][0] = wmma4(a0, b0, acc[0][0]);
          acc[0][1] = wmma4(a0, b1, acc[0][1]);
          acc[1][0] = wmma4(a1, b0, acc[1][0]);
          acc[1][1] = wmma4(a1, b1, acc[1][1]);
        }
        if (rem) {  // K-tail with zero-fill
          const int k = Kmain + hi * 2;
          const bool k0ok = (k < p.Cin), k1ok = ((k + 1) < p.Cin);
          v2f a0, a1, b0, b1;
          a0.x = (vt0 && k0ok) ? p.in[aIdx0 + (long)k * inCst] : 0.f;
          a0.y = (vt0 && k1ok) ? p.in[aIdx0 + (long)(k + 1) * inCst] : 0.f;
          a1.x = (vt1 && k0ok) ? p.in[aIdx1 + (long)k * inCst] : 0.f;
          a1.y = (vt1 && k1ok) ? p.in[aIdx1 + (long)(k + 1) * inCst] : 0.f;
          b0.x = k0ok ? p.w[wIdx0 + (long)k * wCst] : 0.f;
          b0.y = k1ok ? p.w[wIdx0 + (long)(k + 1) * wCst] : 0.f;
          b1.x = k0ok ? p.w[wIdx1 + (long)k * wCst] : 0.f;
          b1.y = k1ok ? p.w[wIdx1 + (long)(k + 1) * wCst] : 0.f;
          acc[0][0] = wmma4(a0, b0, acc[0][0]);
          acc[0][1] = wmma4(a0, b1, acc[0][1]);
          acc[1][0] = wmma4(a1, b0, acc[1][0]);
          acc[1][1] = wmma4(a1, b1, acc[1][1]);
        }
      }

#pragma unroll
  for (int nj = 0; nj < 2; ++nj) {
    const int co = nBase + nj * 16 + l16;
    if (co < p.Cout) {
      const float bb = p.bias ? p.bias[co] : 0.f;
      float sc = 1.f, sh2 = 0.f;
      if (p.bn_g) {
        sc = p.bn_g[co] * 0.9999950000374997f;  // g / sqrt(1 + 1e-5)
        sh2 = p.bn_b[co];
      }
#pragma unroll
      for (int mi = 0; mi < 2; ++mi) {
#pragma unroll
        for (int i = 0; i < 8; ++i) {
          const long m = mBase + mi * 16 + hi * 8 + i;
          if (m < Mtot) {
            float v = acc[mi][nj][i] + bb;
            v = sc * v + sh2;
            if (p.act) v = fmaxf(v, 0.f);
            const long sp = m % spOut;
            const long nb = m / spOut;
            const long oi = ((long)nb * p.Cout + co) * spOut + sp;
            if (p.res) v += p.res[oi];
            p.out[oi] = v;
          }
        }
      }
    }
  }
}

// ---------------------------------------------------------------------
// LayerNorm over last dim (rows x D), one block per row.
// ---------------------------------------------------------------------
__global__ __launch_bounds__(256) void layernorm_kernel(
    const float* __restrict__ x, const float* __restrict__ g,
    const float* __restrict__ b, float* __restrict__ out, int D) {
  const int row = blockIdx.x;
  const float* xr = x + (long)row * D;
  float s = 0.f, s2 = 0.f;
  for (int i = threadIdx.x; i < D; i += 256) {
    const float v = xr[i];
    s += v; s2 += v * v;
  }
  for (int off = 16; off > 0; off >>= 1) {
    s += __shfl_down(s, off, 32);
    s2 += __shfl_down(s2, off, 32);
  }
  __shared__ float rs[8], rs2[8];
  const int wv = threadIdx.x >> 5, ln = threadIdx.x & 31;
  if (ln == 0) { rs[wv] = s; rs2[wv] = s2; }
  __syncthreads();
  if (threadIdx.x == 0) {
    float a = 0.f, a2 = 0.f;
    for (int i = 0; i < 8; ++i) { a += rs[i]; a2 += rs2[i]; }
    rs[0] = a; rs2[0] = a2;
  }
  __syncthreads();
  const float mu = rs[0] / D;
  const float var = rs2[0] / D - mu * mu;
  const float inv = rsqrtf(var + 1e-5f);
  for (int i = threadIdx.x; i < D; i += 256)
    out[(long)row * D + i] = (xr[i] - mu) * inv * g[i] + b[i];
}

// ---------------------------------------------------------------------
// Zero a small scratch region (backs the stride-0 dummy loads).
// ---------------------------------------------------------------------
__global__ void zero_kernel(float* __restrict__ p, int n) {
  const int i = blockIdx.x * blockDim.x + threadIdx.x;
  if (i < n) p[i] = 0.f;
}

// ---------------------------------------------------------------------
// Transposes between (B,C,L) and (B,L,C)
// ---------------------------------------------------------------------
__global__ __launch_bounds__(256) void t_bcl_to_blc(
    const float* __restrict__ in, float* __restrict__ out,
    int Bn, int C, int L) {
  const long t = (long)blockIdx.x * blockDim.x + threadIdx.x;
  if (t >= (long)Bn * C * L) return;
  const int l = (int)(t % L);
  const long r = t / L;
  const int c = (int)(r % C);
  const int b = (int)(r / C);
  out[((long)b * L + l) * C + c] = in[t];
}

__global__ __launch_bounds__(256) void t_blc_to_bcl(
    const float* __restrict__ in, float* __restrict__ out,
    int Bn, int C, int L) {
  const long t = (long)blockIdx.x * blockDim.x + threadIdx.x;
  if (t >= (long)Bn * C * L) return;
  const int c = (int)(t % C);
  const long r = t / C;
  const int l = (int)(r % L);
  const int b = (int)(r / L);
  out[((long)b * C + c) * L + l] = in[t];
}

// ---------------------------------------------------------------------
// Causal depthwise conv1d (k=4, left pad 3) + bias + SiLU over the first
// di features of xz (B,L,2*di) -> xconv (B,L,di)
// ---------------------------------------------------------------------
__global__ __launch_bounds__(256) void dwconv_silu_kernel(
    const float* __restrict__ xz, const float* __restrict__ cw,
    const float* __restrict__ cb, float* __restrict__ xconv,
    int Bn, int L, int di) {
  const long t = (long)blockIdx.x * blockDim.x + threadIdx.x;
  if (t >= (long)Bn * L * di) return;
  const int d = (int)(t % di);
  const long bl = t / di;
  const int l = (int)(bl % L);
  const int b = (int)(bl / L);
  float acc = cb[d];
#pragma unroll
  for (int j = 0; j < 4; ++j) {
    const int li = l - 3 + j;
    if (li >= 0) acc += xz[((long)b * L + li) * (2 * di) + d] * cw[d * 4 + j];
  }
  xconv[t] = acc / (1.f + __expf(-acc));  // silu
}

// ---------------------------------------------------------------------
// Mamba selective scan: one thread per (b, channel d).
// ---------------------------------------------------------------------
template <int S>
__global__ __launch_bounds__(256) void mamba_scan_kernel(
    const float* __restrict__ delta,  // B,L,di
    const float* __restrict__ dbl,    // B,L,(R+2S)
    int R,
    const float* __restrict__ xconv,  // B,L,di
    const float* __restrict__ xz,     // B,L,2di (z at +di)
    const float* __restrict__ A_log,  // di,S
    const float* __restrict__ Dp,     // di
    float* __restrict__ y,            // B,L,di
    int Bn, int L, int di) {
  const int t = blockIdx.x * blockDim.x + threadIdx.x;
  if (t >= Bn * di) return;
  const int b = t / di, d = t % di;
  float Ar[S], h[S];
#pragma unroll
  for (int n = 0; n < S; ++n) {
    Ar[n] = -__expf(A_log[d * S + n]);
    h[n] = 0.f;
  }
  const float Dd = Dp[d];
  const int ldd = R + 2 * S;
  for (int l = 0; l < L; ++l) {
    const long row = (long)b * L + l;
    const float dt = delta[row * di + d];
    const float xt = xconv[row * di + d];
    const float* bc = dbl + row * ldd + R;
    float yv = 0.f;
#pragma unroll
    for (int n = 0; n < S; ++n) {
      h[n] = h[n] * __expf(dt * Ar[n]) + dt * bc[n] * xt;
      yv += h[n] * bc[S + n];
    }
    yv += xt * Dd;
    const float z = xz[row * (2 * di) + di + d];
    yv *= z / (1.f + __expf(-z));
    y[row * di + d] = yv;
  }
}

// ---------------------------------------------------------------------
// Final 1x1x1 conv (Cout = 1)
// ---------------------------------------------------------------------
__global__ __launch_bounds__(256) void final_conv_kernel(
    const float* __restrict__ x, const float* __restrict__ fw,
    const float* __restrict__ fb, float* __restrict__ out,
    int Bn, int C, int SP) {
  const long t = (long)blockIdx.x * blockDim.x + threadIdx.x;
  if (t >= (long)Bn * SP) return;
  const int sp = (int)(t % SP);
  const int b = (int)(t / SP);
  float acc = fb[0];
  for (int c = 0; c < C; ++c) acc += x[((long)b * C + c) * SP + sp] * fw[c];
  out[t] = acc;
}

// =====================================================================
// Host side
// =====================================================================
static inline long cdiv(long a, long b) { return (a + b - 1) / b; }

static void launch_gemm(const float* A, int lda, const float* Bw, int ldb,
                        const float* bias, float* out, int ldo,
                        long M, int N, int K, int act, hipStream_t s) {
  dim3 grid((unsigned)cdiv(M, 128), (unsigned)cdiv(N, 32));
  gemm_wmma_kernel<<<grid, 128, 0, s>>>(A, lda, Bw, ldb, bias, out, ldo,
                                        (int)M, N, K, act);
}

struct ConvSpec {
  int KD, KH, KW;
  int sd, sh, sw;
  int pd, ph, pw;
  int dd, dh, dw;
  int wt_flip;
};

static void launch_conv(const float* in, int N, int Cin, int ID, int IH, int IW,
                        const float* w, int Cout, ConvSpec cs,
                        const float* bias, const float* bng, const float* bnb,
                        const float* res, int act, float* out,
                        const float* zbuf, hipStream_t s) {
  const int OD = ((ID - 1) * cs.dd + 1 + 2 * cs.pd - cs.KD) / cs.sd + 1;
  const int OH = ((IH - 1) * cs.dh + 1 + 2 * cs.ph - cs.KH) / cs.sh + 1;
  const int OW = ((IW - 1) * cs.dw + 1 + 2 * cs.pw - cs.KW) / cs.sw + 1;
  Conv3dArgs p;
  p.in = in; p.w = w; p.bias = bias; p.bn_g = bng; p.bn_b = bnb; p.res = res;
  p.zbuf = zbuf; p.out = out;
  p.N = N; p.Cin = Cin; p.ID = ID; p.IH = IH; p.IW = IW;
  p.Cout = Cout; p.OD = OD; p.OH = OH; p.OW = OW;
  p.KD = cs.KD; p.KH = cs.KH; p.KW = cs.KW;
  p.sd = cs.sd; p.sh = cs.sh; p.sw = cs.sw;
  p.pd = cs.pd; p.ph = cs.ph; p.pw = cs.pw;
  p.dd = cs.dd; p.dh = cs.dh; p.dw = cs.dw;
  p.wt_flip = cs.wt_flip; p.act = act;
  const long Mtot = (long)N * OD * OH * OW;
  dim3 grid((unsigned)cdiv(Mtot, 128), (unsigned)cdiv(Cout, 32));
  conv3d_wmma_kernel<<<grid, 128, 0, s>>>(p);
}

static void run_mamba(const float* x_bcl, float* out_bcl, int Bn, int C, int L,
                      int di, int S, int R, const float* Alog, const float* Dp,
                      const float* convb, const float* convw, const float* dtb,
                      const float* dtw, const float* inproj,
                      const float* outproj, const float* xproj, float* XT,
                      float* XZ, float* XC, float* DBL, float* DEL, float* YB,
                      hipStream_t s) {
  const long BL = (long)Bn * L;
  const long tot = BL * C;
  t_bcl_to_blc<<<(unsigned)cdiv(tot, 256), 256, 0, s>>>(x_bcl, XT, Bn, C, L);
  // in_proj: (BL,C)x(2di,C)^T -> XZ (BL, 2di)
  launch_gemm(XT, C, inproj, C, nullptr, XZ, 2 * di, BL, 2 * di, C, 0, s);
  // depthwise causal conv + silu -> XC (BL, di)
  dwconv_silu_kernel<<<(unsigned)cdiv(BL * di, 256), 256, 0, s>>>(
      XZ, convw, convb, XC, Bn, L, di);
  // x_proj -> DBL (BL, R+2S)
  const int E = R + 2 * S;
  launch_gemm(XC, di, xproj, di, nullptr, DBL, E, BL, E, di, 0, s);
  // delta = softplus(dt @ dt_w^T + dt_b) -> DEL (BL, di)
  launch_gemm(DBL, E, dtw, R, dtb, DEL, di, BL, di, R, 2, s);
  // scan -> YB (BL, di)
  const unsigned sb = (unsigned)cdiv((long)Bn * di, 256);
  if (S == 16)
    mamba_scan_kernel<16><<<sb, 256, 0, s>>>(DEL, DBL, R, XC, XZ, Alog, Dp, YB,
                                             Bn, L, di);
  else if (S == 8)
    mamba_scan_kernel<8><<<sb, 256, 0, s>>>(DEL, DBL, R, XC, XZ, Alog, Dp, YB,
                                            Bn, L, di);
  else
    mamba_scan_kernel<4><<<sb, 256, 0, s>>>(DEL, DBL, R, XC, XZ, Alog, Dp, YB,
                                            Bn, L, di);
  // out_proj -> XT (BL, C)
  launch_gemm(YB, di, outproj, di, nullptr, XT, C, BL, C, di, 0, s);
  t_blc_to_bcl<<<(unsigned)cdiv(tot, 256), 256, 0, s>>>(XT, out_bcl, Bn, C, L);
}

// Input index map (x first, then params flattened JAX-pytree style).
enum {
  IN_X = 0,
  IN_BN1_B, IN_BN1_G, IN_BN2_B, IN_BN2_G, IN_BN3_B, IN_BN3_G,
  IN_F_B, IN_F_W,
  IN_L1R0_W1, IN_L1R0_W2, IN_L1R1_W1, IN_L1R1_W2, IN_L1R2_W1, IN_L1R2_W2,
  IN_L1_W1, IN_L1_W2,
  IN_L2R_B1, IN_L2R_B2, IN_L2R_B3, IN_L2R_W1, IN_L2R_W2, IN_L2R_W3,
  IN_L2_W1, IN_L2_W2,
  IN_L3R_B1, IN_L3R_B2, IN_L3R_B3, IN_L3R_W1, IN_L3R_W2, IN_L3R_W3,
  IN_L3_W1, IN_L3_W2,
  IN_L4R_B1, IN_L4R_B2, IN_L4R_B3, IN_L4R_W1, IN_L4R_W2, IN_L4R_W3,
  IN_L4_W1, IN_L4_W2,
  IN_LN_B, IN_LN_G,
  IN_M1_ALOG, IN_M1_D, IN_M1_CONVB, IN_M1_CONVW, IN_M1_DTB, IN_M1_DTW,
  IN_M1_INPROJ, IN_M1_OUTPROJ, IN_M1_XPROJ,
  IN_M2_ALOG, IN_M2_D, IN_M2_CONVB, IN_M2_CONVW, IN_M2_DTB, IN_M2_DTW,
  IN_M2_INPROJ, IN_M2_OUTPROJ, IN_M2_XPROJ,
  IN_M3_ALOG, IN_M3_D, IN_M3_CONVB, IN_M3_CONVW, IN_M3_DTB, IN_M3_DTW,
  IN_M3_INPROJ, IN_M3_OUTPROJ, IN_M3_XPROJ,
  IN_UP1_B, IN_UP1_W, IN_UP2_B, IN_UP2_W, IN_UP3_B, IN_UP3_W,
  IN_COUNT
};

extern "C" void kernel_launch(void* const* d_in, const int* in_sizes, int n_in,
                              void* d_out, int out_size, void* d_ws,
                              size_t ws_size, hipStream_t stream) {
  (void)in_sizes; (void)n_in; (void)out_size; (void)ws_size;
  auto F = [&](int i) { return (const float*)d_in[i]; };
  float* ws = (float*)d_ws;
  size_t off = 0;
  auto alloc = [&](size_t n) { float* p = ws + off; off += n; return p; };

  float* ZB = alloc(64);        // zero line for stride-0 dummy loads
  float* A0 = alloc(4194304);   // 5D activation ping
  float* A1 = alloc(4194304);   // 5D activation pong
  float* A2 = alloc(4194304);   // residual holder
  float* XT = alloc(4194304);   // (B,L,C) transposed view
  float* XZ = alloc(33554432);  // mamba in_proj output
  float* XC = alloc(16777216);  // mamba conv output
  float* DBL = alloc(1310720);  // mamba x_proj output
  float* DEL = alloc(16777216); // mamba delta
  float* YB = alloc(16777216);  // mamba scan output

  const int B = 4;

  zero_kernel<<<1, 64, 0, stream>>>(ZB, 64);

  // ---- LayerNorm (4,49,1024) -> reshape (4,784,4,4,4) in A0 ----
  layernorm_kernel<<<196, 256, 0, stream>>>(F(IN_X), F(IN_LN_G), F(IN_LN_B),
                                            A0, 1024);

  // ---- l1: basic_r2p1 (stride 1, pad 1) ----
  ConvSpec cs;
  cs = {1, 3, 3, 1, 1, 1, 0, 1, 1, 1, 1, 1, 0};
  launch_conv(A0, B, 784, 4, 4, 4, F(IN_L1_W1), 294, cs, nullptr, nullptr,
              nullptr, nullptr, 1, A1, ZB, stream);
  cs = {3, 1, 1, 1, 1, 1, 1, 0, 0, 1, 1, 1, 0};
  launch_conv(A1, B, 294, 4, 4, 4, F(IN_L1_W2), 294, cs, nullptr, nullptr,
              nullptr, nullptr, 0, A2, ZB, stream);

  // ---- resblock: 3x basic_r2p1(stride 2, pad 3) + residual A2 ----
  const int r2p1w1[3] = {IN_L1R0_W1, IN_L1R1_W1, IN_L1R2_W1};
  const int r2p1w2[3] = {IN_L1R0_W2, IN_L1R1_W2, IN_L1R2_W2};
  const float* cur = A2;
  for (int i = 0; i < 3; ++i) {
    cs = {1, 4, 4, 1, 2, 2, 0, 3, 3, 1, 1, 1, 0};
    launch_conv(cur, B, 294, 4, 4, 4, F(r2p1w1[i]), 294, cs, nullptr, nullptr,
                nullptr, nullptr, 1, A0, ZB, stream);
    cs = {3, 1, 1, 1, 1, 1, 1, 0, 0, 1, 1, 1, 0};
    const int act = (i < 2) ? 1 : 0;
    const float* res = (i == 2) ? A2 : nullptr;
    launch_conv(A0, B, 294, 4, 4, 4, F(r2p1w2[i]), 294, cs, nullptr, nullptr,
                nullptr, res, act, A1, ZB, stream);
    cur = A1;
  }

  // ---- up1: convT(294->147, k4, s2 via lhs-dil) + bias + BN + relu ----
  cs = {4, 4, 4, 1, 1, 1, 2, 2, 2, 2, 2, 2, 1};
  launch_conv(A1, B, 294, 4, 4, 4, F(IN_UP1_W), 147, cs, F(IN_UP1_B),
              F(IN_BN1_G), F(IN_BN1_B), nullptr, 1, A0, ZB,
              stream);  // -> (4,147,8,8,8)

  // ---- l2: basic_r2p1 + resblock2(147) ----
  cs = {1, 3, 3, 1, 1, 1, 0, 1, 1, 1, 1, 1, 0};
  launch_conv(A0, B, 147, 8, 8, 8, F(IN_L2_W1), 147, cs, nullptr, nullptr,
              nullptr, nullptr, 1, A1, ZB, stream);
  cs = {3, 1, 1, 1, 1, 1, 1, 0, 0, 1, 1, 1, 0};
  launch_conv(A1, B, 147, 8, 8, 8, F(IN_L2_W2), 147, cs, nullptr, nullptr,
              nullptr, nullptr, 0, A2, ZB, stream);
  cs = {3, 3, 3, 1, 1, 1, 1, 1, 1, 1, 1, 1, 0};
  launch_conv(A2, B, 147, 8, 8, 8, F(IN_L2R_W1), 147, cs, F(IN_L2R_B1),
              nullptr, nullptr, nullptr, 1, A0, ZB, stream);
  launch_conv(A0, B, 147, 8, 8, 8, F(IN_L2R_W2), 147, cs, F(IN_L2R_B2),
              nullptr, nullptr, nullptr, 1, A1, ZB, stream);
  cs = {1, 1, 1, 1, 1, 1, 0, 0, 0, 1, 1, 1, 0};
  launch_conv(A1, B, 147, 8, 8, 8, F(IN_L2R_W3), 147, cs, F(IN_L2R_B3),
              nullptr, nullptr, A2, 0, A0, ZB, stream);

  // ---- mamba 1: C=147, L=512, di=2352, S=16, R=10 ----
  run_mamba(A0, A0, B, 147, 512, 2352, 16, 10, F(IN_M1_ALOG), F(IN_M1_D),
            F(IN_M1_CONVB), F(IN_M1_CONVW), F(IN_M1_DTB), F(IN_M1_DTW),
            F(IN_M1_INPROJ), F(IN_M1_OUTPROJ), F(IN_M1_XPROJ), XT, XZ, XC, DBL,
            DEL, YB, stream);

  // ---- up2: convT(147->64) + bias + BN + relu -> (4,64,16,16,16) ----
  cs = {4, 4, 4, 1, 1, 1, 2, 2, 2, 2, 2, 2, 1};
  launch_conv(A0, B, 147, 8, 8, 8, F(IN_UP2_W), 64, cs, F(IN_UP2_B),
              F(IN_BN2_G), F(IN_BN2_B), nullptr, 1, A1, ZB, stream);

  // ---- l3: basic_r2p1 + resblock2(64) ----
  cs = {1, 3, 3, 1, 1, 1, 0, 1, 1, 1, 1, 1, 0};
  launch_conv(A1, B, 64, 16, 16, 16, F(IN_L3_W1), 64, cs, nullptr, nullptr,
              nullptr, nullptr, 1, A0, ZB, stream);
  cs = {3, 1, 1, 1, 1, 1, 1, 0, 0, 1, 1, 1, 0};
  launch_conv(A0, B, 64, 16, 16, 16, F(IN_L3_W2), 64, cs, nullptr, nullptr,
              nullptr, nullptr, 0, A2, ZB, stream);
  cs = {3, 3, 3, 1, 1, 1, 1, 1, 1, 1, 1, 1, 0};
  launch_conv(A2, B, 64, 16, 16, 16, F(IN_L3R_W1), 64, cs, F(IN_L3R_B1),
              nullptr, nullptr, nullptr, 1, A0, ZB, stream);
  launch_conv(A0, B, 64, 16, 16, 16, F(IN_L3R_W2), 64, cs, F(IN_L3R_B2),
              nullptr, nullptr, nullptr, 1, A1, ZB, stream);
  cs = {1, 1, 1, 1, 1, 1, 0, 0, 0, 1, 1, 1, 0};
  launch_conv(A1, B, 64, 16, 16, 16, F(IN_L3R_W3), 64, cs, F(IN_L3R_B3),
              nullptr, nullptr, A2, 0, A0, ZB, stream);

  // ---- mamba 2: C=64, L=4096, di=512, S=8, R=4 ----
  run_mamba(A0, A0, B, 64, 4096, 512, 8, 4, F(IN_M2_ALOG), F(IN_M2_D),
            F(IN_M2_CONVB), F(IN_M2_CONVW), F(IN_M2_DTB), F(IN_M2_DTW),
            F(IN_M2_INPROJ), F(IN_M2_OUTPROJ), F(IN_M2_XPROJ), XT, XZ, XC, DBL,
            DEL, YB, stream);

  // ---- up3: convT(64->32) + bias + BN + relu -> (4,32,32,32,32) ----
  cs = {4, 4, 4, 1, 1, 1, 2, 2, 2, 2, 2, 2, 1};
  launch_conv(A0, B, 64, 16, 16, 16, F(IN_UP3_W), 32, cs, F(IN_UP3_B),
              F(IN_BN3_G), F(IN_BN3_B), nullptr, 1, A1, ZB, stream);

  // ---- l4: basic_r2p1 + resblock2(32) ----
  cs = {1, 3, 3, 1, 1, 1, 0, 1, 1, 1, 1, 1, 0};
  launch_conv(A1, B, 32, 32, 32, 32, F(IN_L4_W1), 32, cs, nullptr, nullptr,
              nullptr, nullptr, 1, A0, ZB, stream);
  cs = {3, 1, 1, 1, 1, 1, 1, 0, 0, 1, 1, 1, 0};
  launch_conv(A0, B, 32, 32, 32, 32, F(IN_L4_W2), 32, cs, nullptr, nullptr,
              nullptr, nullptr, 0, A2, ZB, stream);
  cs = {3, 3, 3, 1, 1, 1, 1, 1, 1, 1, 1, 1, 0};
  launch_conv(A2, B, 32, 32, 32, 32, F(IN_L4R_W1), 32, cs, F(IN_L4R_B1),
              nullptr, nullptr, nullptr, 1, A0, ZB, stream);
  launch_conv(A0, B, 32, 32, 32, 32, F(IN_L4R_W2), 32, cs, F(IN_L4R_B2),
              nullptr, nullptr, nullptr, 1, A1, ZB, stream);
  cs = {1, 1, 1, 1, 1, 1, 0, 0, 0, 1, 1, 1, 0};
  launch_conv(A1, B, 32, 32, 32, 32, F(IN_L4R_W3), 32, cs, F(IN_L4R_B3),
              nullptr, nullptr, A2, 0, A0, ZB, stream);

  // ---- mamba 3: C=32, L=32768, di=128, S=4, R=2 ----
  run_mamba(A0, A0, B, 32, 32768, 128, 4, 2, F(IN_M3_ALOG), F(IN_M3_D),
            F(IN_M3_CONVB), F(IN_M3_CONVW), F(IN_M3_DTB), F(IN_M3_DTW),
            F(IN_M3_INPROJ), F(IN_M3_OUTPROJ), F(IN_M3_XPROJ), XT, XZ, XC, DBL,
            DEL, YB, stream);

  // ---- final 1x1x1 conv (32 -> 1) ----
  const long SP = 32768;
  final_conv_kernel<<<(unsigned)cdiv(B * SP, 256), 256, 0, stream>>>(
      A0, F(IN_F_W), F(IN_F_B), (float*)d_out, B, 32, (int)SP);
}